// Supernode_43267500539966
// MI455X (gfx1250) — compile-verified
//
#include <hip/hip_runtime.h>
#include <hip/hip_bf16.h>

typedef float v2f __attribute__((ext_vector_type(2)));
typedef float v8f __attribute__((ext_vector_type(8)));

#define OUTC 256

// ---------------- small elementwise / degree kernels ----------------

__global__ void fill_ones_kernel(float* __restrict__ p, int n) {
  int i = blockIdx.x * blockDim.x + threadIdx.x;
  if (i < n) p[i] = 1.0f;
}

__global__ void deg_count_kernel(const long long* __restrict__ dst,
                                 float* __restrict__ deg, int e) {
  int i = blockIdx.x * blockDim.x + threadIdx.x;
  if (i < e) atomicAdd(&deg[(int)dst[i]], 1.0f);
}

__global__ void rsqrt_inplace_kernel(float* __restrict__ p, int n) {
  int i = blockIdx.x * blockDim.x + threadIdx.x;
  if (i < n) p[i] = rsqrtf(p[i]);
}

// out[i][c..c+3] (or +=) dinv[i]^2 * P[i][c..c+3] + bias[c..c+3]
template <bool ACC>
__global__ void node_axpb_kernel(const float* __restrict__ dinv,
                                 const float* __restrict__ P,
                                 const float* __restrict__ bias,
                                 float* __restrict__ out, int n) {
  int idx = blockIdx.x * blockDim.x + threadIdx.x;  // n*64 threads, float4 each
  if (idx >= n * 64) return;
  int i = idx >> 6;
  int c = (idx & 63) * 4;
  float d = dinv[i];
  float s = d * d;
  const float4 p  = *(const float4*)(P + (size_t)i * OUTC + c);
  const float4 bv = *(const float4*)(bias + c);
  float4 r;
  r.x = s * p.x + bv.x;
  r.y = s * p.y + bv.y;
  r.z = s * p.z + bv.z;
  r.w = s * p.w + bv.w;
  float* o = out + (size_t)i * OUTC + c;
  if (ACC) {
    float4 ov = *(const float4*)o;
    r.x += ov.x; r.y += ov.y; r.z += ov.z; r.w += ov.w;
  }
  *(float4*)o = r;
}

__global__ void relu4_kernel(float* __restrict__ p, int n4) {
  int i = blockIdx.x * blockDim.x + threadIdx.x;
  if (i >= n4) return;
  float4 v = *(const float4*)(p + (size_t)i * 4);
  v.x = fmaxf(v.x, 0.0f);
  v.y = fmaxf(v.y, 0.0f);
  v.z = fmaxf(v.z, 0.0f);
  v.w = fmaxf(v.w, 0.0f);
  *(float4*)(p + (size_t)i * 4) = v;
}

// ---------------- edge scatter: one wave32 per edge, 8 ch/lane ----------------

__global__ void scatter_edges_kernel(const long long* __restrict__ src,
                                     const long long* __restrict__ dst,
                                     const float* __restrict__ dinv,
                                     const float* __restrict__ P,
                                     float* __restrict__ agg, int e) {
  int w = blockIdx.x * (blockDim.x >> 5) + (threadIdx.x >> 5);
  if (w >= e) return;
  int lane = threadIdx.x & 31;
  int s = (int)src[w];
  int d = (int)dst[w];
  // wave-uniform -> force into SGPRs for scalar address generation
  s = __builtin_amdgcn_readfirstlane(s);
  d = __builtin_amdgcn_readfirstlane(d);
  float nrm = dinv[s] * dinv[d];
  const float* prow = P + (size_t)s * OUTC + lane * 8;
  float* arow = agg + (size_t)d * OUTC + lane * 8;
  float4 a = *(const float4*)(prow);
  float4 b = *(const float4*)(prow + 4);
  atomicAdd(arow + 0, a.x * nrm);
  atomicAdd(arow + 1, a.y * nrm);
  atomicAdd(arow + 2, a.z * nrm);
  atomicAdd(arow + 3, a.w * nrm);
  atomicAdd(arow + 4, b.x * nrm);
  atomicAdd(arow + 5, b.y * nrm);
  atomicAdd(arow + 6, b.z * nrm);
  atomicAdd(arow + 7, b.w * nrm);
}

// ---------------- WMMA f32 GEMM: C[M x 256] (+)= A[M x KDIM] @ W[KDIM x 256] ----------------
// Block = 128 threads (4 waves). Block covers 16 rows x 256 cols.
// Wave `wv` covers cols [wv*64, wv*64+64) as 4 WMMA 16x16 accumulators.
// VGPR layouts per CDNA5 ISA 7.12.2:
//   A 16x4 f32 : lane L -> row (L&15), v2f = A[row][k + 2*(L>>4)], A[row][k+1+2*(L>>4)]
//   B 4x16 f32 : lane L -> col (L&15), v2f = W[k + 2*(L>>4)][col], W[k+1+2*(L>>4)][col]
//   C 16x16    : v8f elem j -> C[mbase + j + 8*(L>>4)][nbase + (L&15)]
template <int KDIM, bool ACC, bool HASBIAS>
__global__ __launch_bounds__(128) void gemm_wmma_kernel(
    const float* __restrict__ A, const float* __restrict__ W,
    float* __restrict__ C, const float* __restrict__ bias, int M) {
  const int wv    = threadIdx.x >> 5;
  const int lane  = threadIdx.x & 31;
  const int lhalf = lane >> 4;
  const int lm    = lane & 15;
  const int mbase = blockIdx.x * 16;
  const int nwave = wv * 64;

  int arow = mbase + lm;
  if (arow >= M) arow = M - 1;  // clamp (M is a multiple of 16 here anyway)

  v8f acc[4];
#pragma unroll
  for (int t = 0; t < 4; ++t) {
    if (ACC) {
#pragma unroll
      for (int j = 0; j < 8; ++j) {
        int row = mbase + j + 8 * lhalf;
        row = row < M ? row : M - 1;
        acc[t][j] = C[(size_t)row * OUTC + nwave + t * 16 + lm];
      }
    } else {
#pragma unroll
      for (int j = 0; j < 8; ++j) acc[t][j] = 0.0f;
    }
  }

  const float* aptr = A + (size_t)arow * KDIM + 2 * lhalf;
#pragma unroll 4
  for (int k = 0; k < KDIM; k += 4) {
    v2f a = *(const v2f*)(aptr + k);
    const int krow = k + 2 * lhalf;
#pragma unroll
    for (int t = 0; t < 4; ++t) {
      const int n = nwave + t * 16 + lm;
      v2f b;
      b.x = W[(size_t)krow * OUTC + n];
      b.y = W[(size_t)(krow + 1) * OUTC + n];
      acc[t] = __builtin_amdgcn_wmma_f32_16x16x4_f32(
          /*neg_a=*/false, a, /*neg_b=*/false, b,
          /*c_mod=*/(short)0, acc[t], /*reuse_a=*/false, /*reuse_b=*/false);
    }
  }

#pragma unroll
  for (int t = 0; t < 4; ++t) {
    const int n = nwave + t * 16 + lm;
    const float bv = HASBIAS ? bias[n] : 0.0f;
#pragma unroll
    for (int j = 0; j < 8; ++j) {
      const int row = mbase + j + 8 * lhalf;
      if (row < M) C[(size_t)row * OUTC + n] = acc[t][j] + bv;
    }
  }
}

// ---------------- host orchestration ----------------

extern "C" void kernel_launch(void* const* d_in, const int* in_sizes, int n_in,
                              void* d_out, int out_size, void* d_ws, size_t ws_size,
                              hipStream_t stream) {
  const float*     x    = (const float*)d_in[0];
  const long long* ei   = (const long long*)d_in[1];  // int64 edge_index [2, E]
  const float*     nbf  = (const float*)d_in[2];
  const float*     ptf  = (const float*)d_in[3];
  const float*     W1   = (const float*)d_in[4];
  const float*     b1   = (const float*)d_in[5];
  const float*     W2   = (const float*)d_in[6];
  const float*     b2   = (const float*)d_in[7];
  const float*     Wn   = (const float*)d_in[8];
  const float*     Wt   = (const float*)d_in[9];
  const float*     btp  = (const float*)d_in[10];
  float*           out  = (float*)d_out;

  const int N = in_sizes[0] / 128;  // 100000
  const int E = in_sizes[1] / 2;    // 1600000
  const long long* src = ei;
  const long long* dst = ei + E;

  // Workspace: dinv[N] | P1[N*256] (later reused as P2) | agg1[N*256]
  float* dinv = (float*)d_ws;
  float* P1   = dinv + (((size_t)N + 255) & ~(size_t)255);
  float* agg1 = P1 + (size_t)N * OUTC;
  float* P2   = P1;  // P1 dead after relu; alias

  const int T = 256;
  const int nodeBlocks  = (N + T - 1) / T;
  const int edgeBlocks  = (E + T - 1) / T;
  const int node4       = N * 64;                 // float4 threads over N x 256
  const int node4Blocks = (node4 + T - 1) / T;
  const int wavesPerBlk = T / 32;
  const int scatBlocks  = (E + wavesPerBlk - 1) / wavesPerBlk;
  const int gemmBlocks  = (N + 15) / 16;

  // 1) degrees (self-loop counts as 1) -> dinv = rsqrt(deg)
  fill_ones_kernel<<<nodeBlocks, T, 0, stream>>>(dinv, N);
  deg_count_kernel<<<edgeBlocks, T, 0, stream>>>(dst, dinv, E);
  rsqrt_inplace_kernel<<<nodeBlocks, T, 0, stream>>>(dinv, N);

  // 2) conv1: P1 = x@W1 ; agg1 = dinv^2*P1 + b1 (self-loop + bias) ; edge scatter ; relu
  gemm_wmma_kernel<128, false, false><<<gemmBlocks, 128, 0, stream>>>(x, W1, P1, nullptr, N);
  node_axpb_kernel<false><<<node4Blocks, T, 0, stream>>>(dinv, P1, b1, agg1, N);
  scatter_edges_kernel<<<scatBlocks, T, 0, stream>>>(src, dst, dinv, P1, agg1, E);
  relu4_kernel<<<node4Blocks, T, 0, stream>>>(agg1, node4);

  // 3) conv2 matmul: P2 = relu(h1) @ W2
  gemm_wmma_kernel<256, false, false><<<gemmBlocks, 128, 0, stream>>>(agg1, W2, P2, nullptr, N);

  // 4) base: out = nbf@Wn ; out += ptf@Wt + btp
  gemm_wmma_kernel<128, false, false><<<gemmBlocks, 128, 0, stream>>>(nbf, Wn, out, nullptr, N);
  gemm_wmma_kernel<256, true, true><<<gemmBlocks, 128, 0, stream>>>(ptf, Wt, out, btp, N);

  // 5) out += dinv^2*P2 + b2 (conv2 self-loop + bias), then edge scatter into out
  node_axpb_kernel<true><<<node4Blocks, T, 0, stream>>>(dinv, P2, b2, out, N);
  scatter_edges_kernel<<<scatBlocks, T, 0, stream>>>(src, dst, dinv, P2, out, E);
}